// GNN_52106543235761
// MI455X (gfx1250) — compile-verified
//
#include <hip/hip_runtime.h>
#include <hip/hip_bf16.h>

typedef __attribute__((ext_vector_type(2))) float v2f;
typedef __attribute__((ext_vector_type(8))) float v8f;

#define THREADS 256

// ---------------------------------------------------------------------------
// Zero the workspace accumulators (harness does not re-poison between replays)
// ---------------------------------------------------------------------------
__global__ void gnn_zero_kernel(unsigned* __restrict__ deg_out,
                                unsigned* __restrict__ deg_in,
                                float* __restrict__ agg, int N) {
    int i = blockIdx.x * blockDim.x + threadIdx.x;
    if (i < N) {
        deg_out[i] = 0u;
        deg_in[i]  = 0u;
        agg[i]     = 0.0f;
    }
}

// ---------------------------------------------------------------------------
// Structural degrees: one u32 atomic per endpoint. Edge index arrays streamed
// as int4 (global_load_b128); counters live in L2 (4 MB each).
// ---------------------------------------------------------------------------
__global__ void gnn_degree_kernel(const int* __restrict__ src,
                                  const int* __restrict__ dst,
                                  unsigned* __restrict__ deg_out,
                                  unsigned* __restrict__ deg_in, int E) {
    int i  = blockIdx.x * blockDim.x + threadIdx.x;
    int E4 = E >> 2;
    if (i < E4) {
        int4 s = reinterpret_cast<const int4*>(src)[i];
        int4 d = reinterpret_cast<const int4*>(dst)[i];
        atomicAdd(&deg_out[s.x], 1u);
        atomicAdd(&deg_out[s.y], 1u);
        atomicAdd(&deg_out[s.z], 1u);
        atomicAdd(&deg_out[s.w], 1u);
        atomicAdd(&deg_in[d.x], 1u);
        atomicAdd(&deg_in[d.y], 1u);
        atomicAdd(&deg_in[d.z], 1u);
        atomicAdd(&deg_in[d.w], 1u);
    }
    if (i < (E & 3)) {                       // scalar tail (E not multiple of 4)
        int e = (E4 << 2) + i;
        atomicAdd(&deg_out[src[e]], 1u);
        atomicAdd(&deg_in[dst[e]], 1u);
    }
}

// ---------------------------------------------------------------------------
// feat[i] = rsqrt(max(deg_out[i],1)) * (pos[i,0]*W0 + pos[i,1]*W1)
// via V_WMMA_F32_16X16X4_F32. One wave handles a 16-node tile:
//   A (16x4 f32): lanes 0-15 hold {K0=pos.x*ns, K1=pos.y*ns}; lanes 16-31 (K2,K3)=0
//   B (4x16 f32): lanes 0-15 hold {K0=W0, K1=W1} for every column; K2,K3 = 0
//   D (16x16)   : every column equals feat; lane 0 holds D[0..7][0] in c[0..7],
//                 lane 16 holds D[8..15][0] -> two lanes store 16 contiguous
//                 feats as two global_store_b128 (scalar fallback on tail tile).
// ---------------------------------------------------------------------------
__global__ void gnn_feat_wmma_kernel(const float* __restrict__ pos,
                                     const unsigned* __restrict__ deg_out,
                                     const float* __restrict__ Wp,
                                     float* __restrict__ feat, int N) {
    int wave = (blockIdx.x * blockDim.x + threadIdx.x) >> 5;
    int lane = threadIdx.x & 31;
    int base = wave << 4;                    // 16 nodes per wave
    if (base >= N) return;                   // whole-wave uniform exit (EXEC stays full)

    v2f a;  a.x = 0.0f;  a.y = 0.0f;
    v2f bm; bm.x = 0.0f; bm.y = 0.0f;
    if (lane < 16) {
        bm.x = Wp[0];                        // B row K=0
        bm.y = Wp[1];                        // B row K=1
        int node = base + lane;
        if (node < N) {
            float dg = (float)deg_out[node];
            float ns = rsqrtf(fmaxf(dg, 1.0f));
            a.x = pos[2 * node]     * ns;    // A K=0
            a.y = pos[2 * node + 1] * ns;    // A K=1
        }
    }
    v8f c = {};
    c = __builtin_amdgcn_wmma_f32_16x16x4_f32(
            /*neg_a=*/false, a, /*neg_b=*/false, bm,
            /*c_mod=*/(short)0, c, /*reuse_a=*/false, /*reuse_b=*/false);

    if ((lane & 15) == 0) {                  // lanes 0 and 16 hold column N=0
        int m0 = base + ((lane >> 4) << 3);  // lane 0 -> M=0..7, lane 16 -> M=8..15
        if (base + 16 <= N) {
            // full tile: two 16-byte stores (m0 is 8-float aligned)
            float4 lo = make_float4(c[0], c[1], c[2], c[3]);
            float4 hi = make_float4(c[4], c[5], c[6], c[7]);
            *reinterpret_cast<float4*>(feat + m0)     = lo;
            *reinterpret_cast<float4*>(feat + m0 + 4) = hi;
        } else {
#pragma unroll
            for (int r = 0; r < 8; ++r) {
                int node = m0 + r;
                if (node < N) feat[node] = c[r];
            }
        }
    }
}

// ---------------------------------------------------------------------------
// Edge scatter: agg[dst] += feat[src] * weight. Streaming b128 loads of
// src/dst/weight; feat gathers + fp32 atomics resolve in L2 (4 MB arrays).
// unsafeAtomicAdd -> hardware global_atomic_add_f32 (no CAS loop, no return).
// ---------------------------------------------------------------------------
__global__ void gnn_scatter_kernel(const int* __restrict__ src,
                                   const int* __restrict__ dst,
                                   const float* __restrict__ w,
                                   const float* __restrict__ feat,
                                   float* __restrict__ agg, int E) {
    int i  = blockIdx.x * blockDim.x + threadIdx.x;
    int E4 = E >> 2;
    if (i < E4) {
        int4   s  = reinterpret_cast<const int4*>(src)[i];
        int4   d  = reinterpret_cast<const int4*>(dst)[i];
        float4 we = reinterpret_cast<const float4*>(w)[i];
        unsafeAtomicAdd(&agg[d.x], feat[s.x] * we.x);
        unsafeAtomicAdd(&agg[d.y], feat[s.y] * we.y);
        unsafeAtomicAdd(&agg[d.z], feat[s.z] * we.z);
        unsafeAtomicAdd(&agg[d.w], feat[s.w] * we.w);
    }
    if (i < (E & 3)) {
        int e = (E4 << 2) + i;
        unsafeAtomicAdd(&agg[dst[e]], feat[src[e]] * w[e]);
    }
}

// ---------------------------------------------------------------------------
// out[i] = agg[i] * rsqrt(max(deg_in[i],1)) + b
// ---------------------------------------------------------------------------
__global__ void gnn_out_kernel(const float* __restrict__ agg,
                               const unsigned* __restrict__ deg_in,
                               const float* __restrict__ bptr,
                               float* __restrict__ out, int N) {
    int i = blockIdx.x * blockDim.x + threadIdx.x;
    if (i < N) {
        float nd = rsqrtf(fmaxf((float)deg_in[i], 1.0f));
        out[i] = agg[i] * nd + bptr[0];
    }
}

extern "C" void kernel_launch(void* const* d_in, const int* in_sizes, int n_in,
                              void* d_out, int out_size, void* d_ws, size_t ws_size,
                              hipStream_t stream) {
    (void)n_in; (void)out_size; (void)ws_size;
    // setup_inputs order: position[N,2] f32, weight[E] f32, src[E] i32,
    //                     dst[E] i32, W[2,1] f32, b[1] f32
    const float* pos = (const float*)d_in[0];
    const float* wgt = (const float*)d_in[1];
    const int*   src = (const int*)d_in[2];
    const int*   dst = (const int*)d_in[3];
    const float* W   = (const float*)d_in[4];
    const float* b   = (const float*)d_in[5];
    float*       out = (float*)d_out;

    const int N = in_sizes[0] / 2;
    const int E = in_sizes[1];

    // Workspace layout: deg_out u32[N] | deg_in u32[N] | feat f32[N] | agg f32[N]
    char* ws = (char*)d_ws;
    unsigned* deg_out = (unsigned*)(ws);
    unsigned* deg_in  = (unsigned*)(ws + (size_t)N * 4);
    float*    feat    = (float*)   (ws + (size_t)N * 8);
    float*    agg     = (float*)   (ws + (size_t)N * 12);

    const int gridN = (N + THREADS - 1) / THREADS;
    const int E4    = (E + 3) / 4;
    const int gridE = (E4 + THREADS - 1) / THREADS;
    const long long waves = ((long long)N + 15) / 16;          // 16 nodes per wave
    const int       gridF = (int)((waves * 32 + THREADS - 1) / THREADS);

    gnn_zero_kernel  <<<gridN, THREADS, 0, stream>>>(deg_out, deg_in, agg, N);
    gnn_degree_kernel<<<gridE, THREADS, 0, stream>>>(src, dst, deg_out, deg_in, E);
    gnn_feat_wmma_kernel<<<gridF, THREADS, 0, stream>>>(pos, deg_out, W, feat, N);
    gnn_scatter_kernel<<<gridE, THREADS, 0, stream>>>(src, dst, wgt, feat, agg, E);
    gnn_out_kernel   <<<gridN, THREADS, 0, stream>>>(agg, deg_in, b, out, N);
}